// TemporalConsistencyLoss_42803644072824
// MI455X (gfx1250) — compile-verified
//
#include <hip/hip_runtime.h>

#define DEVINL __device__ __forceinline__

constexpr int   kB = 4;
constexpr int   kS = 2048;
constexpr int   kN = kB * kS * kS;          // 16,777,216 elements
constexpr float kMargin   = 0.5f;
constexpr float kEps      = 0.1f;
constexpr float kLogClamp = -100.0f;

constexpr int kCausBlocks = 4096;            // 4096*256 threads * 4 iter * 4 elem = kN exactly
constexpr int kNumBlocks  = kCausBlocks + kS;
// ws layout (doubles), every slot rewritten each call (no init kernel needed):
//   [0, kCausBlocks)                        : per-block BCE partial sums
//   [kCausBlocks, kCausBlocks+kS)           : per-row ordering partial sums
//   [kCausBlocks+kS, kCausBlocks+2*kS)      : per-row transitivity partial sums

typedef __attribute__((ext_vector_type(2))) float v2f;
typedef __attribute__((ext_vector_type(8))) float v8f;
typedef __attribute__((ext_vector_type(4))) float f4v;
typedef __attribute__((ext_vector_type(4))) int   i4v;

// ---------------------------------------------------------------------------
// Wave32 lane-sum via V_WMMA_F32_16X16X4_F32 with an all-ones B matrix.
// A (16x4 f32): lanes 0-15 VGPR0 = A[m,0], lanes 16-31 VGPR0 = A[m,2]; VGPR1
// (K=1,3) zeroed. With B==1: D[m,n] = partial[m] + partial[m+16] for every n.
// Summing the 8 D VGPRs per lane gives the half-wave total; xor-shuffle by 16
// merges halves. Exact f32, one matrix op instead of a 5-step butterfly.
// EXEC all-ones required — every call site is block-uniform.
// ---------------------------------------------------------------------------
DEVINL float waveReduceSum(float x) {
#if __has_builtin(__builtin_amdgcn_wmma_f32_16x16x4_f32)
  v2f a;  a[0] = x;    a[1] = 0.0f;
  v2f bo; bo[0] = 1.0f; bo[1] = 1.0f;
  v8f c = {};
  v8f d = __builtin_amdgcn_wmma_f32_16x16x4_f32(
      /*neg_a=*/false, a, /*neg_b=*/false, bo,
      /*c_mod=*/(short)0, c, /*reuse_a=*/false, /*reuse_b=*/false);
  float s = ((d[0] + d[1]) + (d[2] + d[3])) + ((d[4] + d[5]) + (d[6] + d[7]));
  s += __shfl_xor(s, 16, 32);
  return s;
#else
  for (int off = 16; off > 0; off >>= 1) x += __shfl_xor(x, off, 32);
  return x;
#endif
}

// BCE term with torch-style log clamp; t is exactly 0/1 so one log is live.
DEVINL float bceTerm(float p, int t) {
  float lg = (t != 0) ? __logf(p) : __logf(1.0f - p);
  return fmaxf(lg, kLogClamp);
}

// ---------------------------------------------------------------------------
// Fused main kernel. Blocks [0, kCausBlocks) stream the 128 MiB BCE reduction;
// blocks [kCausBlocks, kNumBlocks) each own row i of the S x S ordering +
// transitivity sweep (16 MiB precedence stream + cached time vectors). Both
// memory streams are in flight concurrently -> single pass at HBM rate.
// Per-block result goes to a private ws slot: deterministic, atomic-free.
// ---------------------------------------------------------------------------
__global__ __launch_bounds__(256) void main_kernel(
    const float* __restrict__ pc, const int* __restrict__ tc,
    const float* __restrict__ pt, const float* __restrict__ tt,
    const int* __restrict__ prec, double* __restrict__ ws) {
  __shared__ float wsums[2][8];
  const int wave = threadIdx.x >> 5;
  const int lane = threadIdx.x & 31;

  if (blockIdx.x < (unsigned)kCausBlocks) {
    // ---- causality BCE: exactly 4 float4/int4 pairs per thread, no tail ----
    const f4v* __restrict__ pc4 = (const f4v*)pc;
    const i4v* __restrict__ tc4 = (const i4v*)tc;
    const int base   = blockIdx.x * 256 + threadIdx.x;
    const int stride = kCausBlocks * 256;   // 1,048,576

    float acc = 0.0f;
#pragma unroll
    for (int it = 0; it < 4; ++it) {
      const int idx = base + it * stride;
      f4v p = __builtin_nontemporal_load(pc4 + idx);
      i4v t = __builtin_nontemporal_load(tc4 + idx);
      acc += bceTerm(p[0], t[0]);
      acc += bceTerm(p[1], t[1]);
      acc += bceTerm(p[2], t[2]);
      acc += bceTerm(p[3], t[3]);
    }

    float w = waveReduceSum(acc);
    if (lane == 0) wsums[0][wave] = w;
    __syncthreads();
    if (threadIdx.x == 0) {
      float s = 0.0f;
#pragma unroll
      for (int q = 0; q < 8; ++q) s += wsums[0][q];
      ws[blockIdx.x] = (double)s;
    }
  } else {
    // ---- ordering + transitivity for row i ----
    const int i = blockIdx.x - kCausBlocks;

    float pti[kB], tti[kB];
#pragma unroll
    for (int b = 0; b < kB; ++b) {          // uniform -> scalar loads
      pti[b] = pt[b * kS + i];
      tti[b] = tt[b * kS + i];
    }

    float ord_acc = 0.0f, trans_acc = 0.0f;
    for (int k = threadIdx.x; k < kS; k += 256) {   // exactly 8 uniform iters
      const int   gap  = k - i;
      const float mult = (gap >= 2) ? (float)(gap - 1) : 0.0f;
      const int   pr   = __builtin_nontemporal_load(prec + i * kS + k);

      float vio = 0.0f;
#pragma unroll
      for (int b = 0; b < kB; ++b) {
        const float ptk = pt[b * kS + k];
        const float ttk = tt[b * kS + k];
        const float o   = (tti[b] < ttk) ? 1.0f : 0.0f;
        ord_acc += fmaxf(kMargin - (pti[b] - ptk) * o, 0.0f);
        vio     += fmaxf(kEps - (ptk - pti[b]), 0.0f);
      }
      if (pr > 0) trans_acc += mult * (vio * 0.25f);
    }

    float ow = waveReduceSum(ord_acc);
    float tw = waveReduceSum(trans_acc);
    if (lane == 0) { wsums[0][wave] = ow; wsums[1][wave] = tw; }
    __syncthreads();
    if (threadIdx.x == 0) {
      float so = 0.0f, st = 0.0f;
#pragma unroll
      for (int q = 0; q < 8; ++q) { so += wsums[0][q]; st += wsums[1][q]; }
      ws[kCausBlocks + i]      = (double)so;
      ws[kCausBlocks + kS + i] = (double)st;
    }
  }
}

// ---------------------------------------------------------------------------
// Single-block deterministic f64 tree reduction of the per-block partials,
// then apply means and lambda weights.
// ---------------------------------------------------------------------------
__global__ __launch_bounds__(256) void finalize_kernel(
    const double* __restrict__ ws, float* __restrict__ out) {
  __shared__ double red[3][256];
  double b = 0.0, o = 0.0, tr = 0.0;
  for (int idx = threadIdx.x; idx < kCausBlocks; idx += 256) b += ws[idx];
  for (int idx = threadIdx.x; idx < kS; idx += 256) {
    o  += ws[kCausBlocks + idx];
    tr += ws[kCausBlocks + kS + idx];
  }
  red[0][threadIdx.x] = b;
  red[1][threadIdx.x] = o;
  red[2][threadIdx.x] = tr;
  __syncthreads();
  for (int s = 128; s > 0; s >>= 1) {
    if (threadIdx.x < s) {
      red[0][threadIdx.x] += red[0][threadIdx.x + s];
      red[1][threadIdx.x] += red[1][threadIdx.x + s];
      red[2][threadIdx.x] += red[2][threadIdx.x + s];
    }
    __syncthreads();
  }
  if (threadIdx.x == 0) {
    const double n     = (double)kN;
    const double ord   = red[1][0] / n;
    const double caus  = -(red[0][0] / n);
    const double denom = (double)kS * (double)(kS - 1) * (double)(kS - 2) / 6.0;
    const double trans = red[2][0] / denom;
    out[0] = (float)ord;
    out[1] = (float)caus;
    out[2] = (float)trans;
    out[3] = (float)(1.0 * ord + 0.8 * caus + 0.6 * trans);
  }
}

extern "C" void kernel_launch(void* const* d_in, const int* in_sizes, int n_in,
                              void* d_out, int out_size, void* d_ws, size_t ws_size,
                              hipStream_t stream) {
  (void)in_sizes; (void)n_in; (void)out_size; (void)ws_size;

  const float* pred_times   = (const float*)d_in[0];
  const float* pred_causal  = (const float*)d_in[1];
  const float* target_times = (const float*)d_in[2];
  const int*   target_caus  = (const int*)d_in[3];
  const int*   target_prec  = (const int*)d_in[4];  // only batch 0 is used

  double* ws  = (double*)d_ws;    // 8192 doubles = 64 KB, all rewritten each call
  float*  out = (float*)d_out;

  hipLaunchKernelGGL(main_kernel, dim3(kNumBlocks), dim3(256), 0, stream,
                     pred_causal, target_caus, pred_times, target_times,
                     target_prec, ws);
  hipLaunchKernelGGL(finalize_kernel, dim3(1), dim3(256), 0, stream, ws, out);
}